// GNNModel_18270790877246
// MI455X (gfx1250) — compile-verified
//
#include <hip/hip_runtime.h>

#define N_NODES  100000
#define N_EDGES  3200000
#define N_GRAPHS 1024
#define HID      64
#define TILES    (N_NODES / 16)   // 6250 exactly
#define LN_EPS   1e-5f

typedef __attribute__((ext_vector_type(2))) float v2f;
typedef __attribute__((ext_vector_type(8))) float v8f;

__device__ __forceinline__ v8f wmma4(v2f a, v2f b, v8f c) {
  // D = A(16x4 f32) * B(4x16 f32) + C(16x16 f32), exact f32 path on CDNA5
  return __builtin_amdgcn_wmma_f32_16x16x4_f32(false, a, false, b, (short)0, c, false, false);
}

// ---------------------------------------------------------------- utilities
__global__ void zero_kernel(float* __restrict__ p, int n) {
  int i = blockIdx.x * blockDim.x + threadIdx.x;
  if (i < n) p[i] = 0.f;
}

// ------------------------------------------------------- edge aggregations
__global__ void edge_agg6_kernel(const float* __restrict__ x,
                                 const long long* __restrict__ src,
                                 const long long* __restrict__ dst,
                                 float* __restrict__ agg) {
  long i = (long)blockIdx.x * blockDim.x + threadIdx.x;
  if (i >= (long)N_EDGES * 6) return;
  int e = (int)(i / 6);
  int f = (int)(i - (long)e * 6);
  int s = (int)src[e], d = (int)dst[e];
  atomicAdd(&agg[(long)d * 6 + f], x[(long)s * 6 + f]);
}

__global__ void edge_agg64_kernel(const float* __restrict__ x,
                                  const long long* __restrict__ src,
                                  const long long* __restrict__ dst,
                                  float* __restrict__ agg) {
  long i = (long)blockIdx.x * blockDim.x + threadIdx.x;
  if (i >= (long)N_EDGES * 16) return;
  int e  = (int)(i >> 4);
  int c4 = ((int)i & 15) * 4;
  int s = (int)src[e], d = (int)dst[e];
  const float4 v = *(const float4*)&x[(long)s * 64 + c4];
  float* o = &agg[(long)d * 64 + c4];
  atomicAdd(o + 0, v.x); atomicAdd(o + 1, v.y);
  atomicAdd(o + 2, v.z); atomicAdd(o + 3, v.w);
}

// ----------------------------------------------- GIN layer 1: (x+agg6) 6->64->64
__global__ __launch_bounds__(128)
void gin_mlp6_kernel(const float* __restrict__ x, const float* __restrict__ agg,
                     const float* __restrict__ wa, const float* __restrict__ ba,
                     const float* __restrict__ wb, const float* __restrict__ bb,
                     float* __restrict__ y) {
  __shared__ float WA[8 * 64];      // 6x64 zero-padded to 8x64
  __shared__ float WB[64 * 64];
  __shared__ float T1[4][16 * 8];   // per-wave input tile (K padded to 8)
  __shared__ float HT[4][16 * 64];  // per-wave hidden tile

  int tid = threadIdx.x;
  for (int i = tid; i < 8 * 64; i += 128) WA[i] = ((i >> 6) < 6) ? wa[i] : 0.f;
  for (int i = tid; i < 64 * 64; i += 128) WB[i] = wb[i];
  __syncthreads();

  int wave = tid >> 5, lane = tid & 31;
  int gtile = blockIdx.x * 4 + wave;
  int tile  = (gtile < TILES) ? gtile : (TILES - 1);   // clamp: keep EXEC full for WMMA
  float* T = T1[wave];
  float* H = HT[wave];

  for (int i = lane; i < 16 * 8; i += 32) {
    int m = i >> 3, k = i & 7;
    long node = (long)tile * 16 + m;
    T[i] = (k < 6) ? (x[node * 6 + k] + agg[node * 6 + k]) : 0.f;
  }

  int m     = lane & 15;           // A-row / B-col / C-col for this lane
  int khalf = (lane >> 4) << 1;    // k offset 0 or 2 within a k-step of 4
  int mg    = (lane >> 4) << 3;    // C-row base 0 or 8

  // GEMM1: H = relu(T @ WA + ba)  (K = 8, 2 wmma k-steps)
  v8f acc[4];
  for (int nt = 0; nt < 4; ++nt) {
    float bv = ba[nt * 16 + m];
    v8f c; for (int r = 0; r < 8; ++r) c[r] = bv;
    acc[nt] = c;
  }
  for (int kk = 0; kk < 2; ++kk) {
    int kb = kk * 4 + khalf;
    v2f a; a[0] = T[m * 8 + kb]; a[1] = T[m * 8 + kb + 1];
    for (int nt = 0; nt < 4; ++nt) {
      v2f b; b[0] = WA[kb * 64 + nt * 16 + m]; b[1] = WA[(kb + 1) * 64 + nt * 16 + m];
      acc[nt] = wmma4(a, b, acc[nt]);
    }
  }
  for (int nt = 0; nt < 4; ++nt)
    for (int r = 0; r < 8; ++r) {
      float v = acc[nt][r];
      H[(mg + r) * 64 + nt * 16 + m] = v > 0.f ? v : 0.f;
    }

  // GEMM2: Y = H @ WB + bb  (K = 64, 16 wmma k-steps)
  v8f acc2[4];
  for (int nt = 0; nt < 4; ++nt) {
    float bv = bb[nt * 16 + m];
    v8f c; for (int r = 0; r < 8; ++r) c[r] = bv;
    acc2[nt] = c;
  }
  for (int kk = 0; kk < 16; ++kk) {
    int kb = kk * 4 + khalf;
    v2f a; a[0] = H[m * 64 + kb]; a[1] = H[m * 64 + kb + 1];
    for (int nt = 0; nt < 4; ++nt) {
      v2f b; b[0] = WB[kb * 64 + nt * 16 + m]; b[1] = WB[(kb + 1) * 64 + nt * 16 + m];
      acc2[nt] = wmma4(a, b, acc2[nt]);
    }
  }
  if (gtile < TILES) {
    long base = (long)tile * 16 * 64;
    for (int nt = 0; nt < 4; ++nt)
      for (int r = 0; r < 8; ++r)
        y[base + (mg + r) * 64 + nt * 16 + m] = acc2[nt][r];
  }
}

// ----------------------------------------------- GIN layer 2: (x+agg) 64->64->64
__global__ __launch_bounds__(128)
void gin_mlp64_kernel(const float* x, const float* agg,           // no restrict: y may alias x
                      const float* __restrict__ wa, const float* __restrict__ ba,
                      const float* __restrict__ wb, const float* __restrict__ bb,
                      float* y) {
  __shared__ float WA[64 * 64];
  __shared__ float WB[64 * 64];
  __shared__ float TT[4][16 * 64];  // per-wave tile, reused T -> H

  int tid = threadIdx.x;
  for (int i = tid; i < 64 * 64; i += 128) { WA[i] = wa[i]; WB[i] = wb[i]; }
  __syncthreads();

  int wave = tid >> 5, lane = tid & 31;
  int gtile = blockIdx.x * 4 + wave;
  int tile  = (gtile < TILES) ? gtile : (TILES - 1);
  float* T = TT[wave];
  long base = (long)tile * 16 * 64;

  for (int i = lane; i < 16 * 64; i += 32) T[i] = x[base + i] + agg[base + i];

  int m     = lane & 15;
  int khalf = (lane >> 4) << 1;
  int mg    = (lane >> 4) << 3;

  // GEMM1: H = relu(T @ WA + ba)
  v8f acc[4];
  for (int nt = 0; nt < 4; ++nt) {
    float bv = ba[nt * 16 + m];
    v8f c; for (int r = 0; r < 8; ++r) c[r] = bv;
    acc[nt] = c;
  }
  for (int kk = 0; kk < 16; ++kk) {
    int kb = kk * 4 + khalf;
    v2f a; a[0] = T[m * 64 + kb]; a[1] = T[m * 64 + kb + 1];
    for (int nt = 0; nt < 4; ++nt) {
      v2f b; b[0] = WA[kb * 64 + nt * 16 + m]; b[1] = WA[(kb + 1) * 64 + nt * 16 + m];
      acc[nt] = wmma4(a, b, acc[nt]);
    }
  }
  // same-wave LDS ops are in-order: safe to overwrite T with H
  for (int nt = 0; nt < 4; ++nt)
    for (int r = 0; r < 8; ++r) {
      float v = acc[nt][r];
      T[(mg + r) * 64 + nt * 16 + m] = v > 0.f ? v : 0.f;
    }

  // GEMM2: Y = H @ WB + bb
  v8f acc2[4];
  for (int nt = 0; nt < 4; ++nt) {
    float bv = bb[nt * 16 + m];
    v8f c; for (int r = 0; r < 8; ++r) c[r] = bv;
    acc2[nt] = c;
  }
  for (int kk = 0; kk < 16; ++kk) {
    int kb = kk * 4 + khalf;
    v2f a; a[0] = T[m * 64 + kb]; a[1] = T[m * 64 + kb + 1];
    for (int nt = 0; nt < 4; ++nt) {
      v2f b; b[0] = WB[kb * 64 + nt * 16 + m]; b[1] = WB[(kb + 1) * 64 + nt * 16 + m];
      acc2[nt] = wmma4(a, b, acc2[nt]);
    }
  }
  if (gtile < TILES) {
    for (int nt = 0; nt < 4; ++nt)
      for (int r = 0; r < 8; ++r)
        y[base + (mg + r) * 64 + nt * 16 + m] = acc2[nt][r];
  }
}

// ----------------------------------------------------------- graph LayerNorm
__global__ void stats1_kernel(const float* __restrict__ x, const long long* __restrict__ batch,
                              float* __restrict__ gsum, float* __restrict__ gcnt, int do_cnt) {
  int n = blockIdx.x * blockDim.x + threadIdx.x;
  if (n >= N_NODES) return;
  const float* row = &x[(long)n * 64];
  float s = 0.f;
  for (int c = 0; c < 64; ++c) s += row[c];
  int g = (int)batch[n];
  atomicAdd(&gsum[g], s);
  if (do_cnt) atomicAdd(&gcnt[g], 1.f);
}

__global__ void ln_mean_kernel(const float* __restrict__ gsum, const float* __restrict__ gcnt,
                               float* __restrict__ gmean) {
  int g = blockIdx.x * blockDim.x + threadIdx.x;
  if (g >= N_GRAPHS) return;
  float c = gcnt[g]; if (c < 1.f) c = 1.f;
  gmean[g] = gsum[g] / (c * 64.f);
}

__global__ void stats2_kernel(const float* __restrict__ x, const long long* __restrict__ batch,
                              const float* __restrict__ gmean, float* __restrict__ gvar) {
  int n = blockIdx.x * blockDim.x + threadIdx.x;
  if (n >= N_NODES) return;
  int g = (int)batch[n];
  float mu = gmean[g];
  const float* row = &x[(long)n * 64];
  float s = 0.f;
  for (int c = 0; c < 64; ++c) { float d = row[c] - mu; s += d * d; }
  atomicAdd(&gvar[g], s);
}

__global__ void ln_rsig_kernel(const float* __restrict__ gvar, const float* __restrict__ gcnt,
                               float* __restrict__ grsig) {
  int g = blockIdx.x * blockDim.x + threadIdx.x;
  if (g >= N_GRAPHS) return;
  float c = gcnt[g]; if (c < 1.f) c = 1.f;
  grsig[g] = rsqrtf(gvar[g] / (c * 64.f) + LN_EPS);
}

__global__ void ln_apply_relu_kernel(float* __restrict__ x, const long long* __restrict__ batch,
                                     const float* __restrict__ gmean, const float* __restrict__ grsig,
                                     const float* __restrict__ w, const float* __restrict__ b) {
  long i = (long)blockIdx.x * blockDim.x + threadIdx.x;
  if (i >= (long)N_NODES * 64) return;
  int n = (int)(i >> 6), c = (int)i & 63;
  int g = (int)batch[n];
  float v = (x[i] - gmean[g]) * grsig[g] * w[c] + b[c];
  x[i] = v > 0.f ? v : 0.f;
}

// ----------------------------------------------------------------- pooling
__global__ void pool_kernel(const float* __restrict__ x, const long long* __restrict__ batch,
                            float* __restrict__ gadd) {
  long i = (long)blockIdx.x * blockDim.x + threadIdx.x;
  if (i >= (long)N_NODES * 64) return;
  int n = (int)(i >> 6), c = (int)i & 63;
  atomicAdd(&gadd[(long)batch[n] * 64 + c], x[i]);
}

__global__ void pool_finish_kernel(const float* __restrict__ gadd, const float* __restrict__ gcnt,
                                   float* __restrict__ emb) {
  int i = blockIdx.x * blockDim.x + threadIdx.x;
  if (i >= N_GRAPHS * 64) return;
  int g = i >> 6;
  float c = gcnt[g]; if (c < 1.f) c = 1.f;
  emb[i] = gadd[i] * (1.f + 1.f / c);
}

// -------------------------------------------------------------------- head
__global__ void fc1_kernel(const float* __restrict__ emb1, const float* __restrict__ emb2,
                           const float* __restrict__ d1, const float* __restrict__ d2,
                           const float* __restrict__ w, const float* __restrict__ b,
                           float* __restrict__ h1) {
  int i = blockIdx.x * blockDim.x + threadIdx.x;
  if (i >= N_GRAPHS * 128) return;
  int g = i >> 7, o = i & 127;
  float acc = b[o];
  for (int k = 0; k < 64; ++k) acc += emb1[g * 64 + k] * w[k * 128 + o];
  for (int k = 0; k < 64; ++k) acc += emb2[g * 64 + k] * w[(64 + k) * 128 + o];
  for (int k = 0; k < 5;  ++k) acc += d1[g * 5 + k]  * w[(128 + k) * 128 + o];
  for (int k = 0; k < 5;  ++k) acc += d2[g * 5 + k]  * w[(133 + k) * 128 + o];
  h1[i] = acc > 0.f ? acc : 0.f;
}

__global__ void fc2_kernel(const float* __restrict__ h1, const float* __restrict__ w,
                           const float* __restrict__ b, float* __restrict__ h2) {
  int i = blockIdx.x * blockDim.x + threadIdx.x;
  if (i >= N_GRAPHS * 64) return;
  int g = i >> 6, o = i & 63;
  float acc = b[o];
  for (int k = 0; k < 128; ++k) acc += h1[g * 128 + k] * w[k * 64 + o];
  h2[i] = acc > 0.f ? acc : 0.f;
}

__global__ void out_kernel(const float* __restrict__ h2, const float* __restrict__ w,
                           const float* __restrict__ b, float* __restrict__ out) {
  int g = blockIdx.x * blockDim.x + threadIdx.x;
  if (g >= N_GRAPHS) return;
  float acc = b[0];
  for (int k = 0; k < 64; ++k) acc += h2[g * 64 + k] * w[k];
  out[g] = acc;
}

// -------------------------------------------------------------- host driver
namespace {
struct Scratch {
  float *B0, *B1, *AGG6, *GCNT, *GSUM, *GVAR, *GMEAN, *GRSIG, *GADD;
};

inline void launch_zero(float* p, int n, hipStream_t s) {
  zero_kernel<<<(n + 255) / 256, 256, 0, s>>>(p, n);
}

void run_backbone(const float* x, const long long* ei, const long long* batch,
                  const float* w11, const float* b11, const float* w12, const float* b12,
                  const float* ln1w, const float* ln1b,
                  const float* w21, const float* b21, const float* w22, const float* b22,
                  const float* ln2w, const float* ln2b,
                  float* emb, const Scratch& S, hipStream_t st) {
  const long long* src = ei;
  const long long* dst = ei + N_EDGES;
  const int nodeBlocks = (N_NODES + 255) / 256;
  const int ncBlocks   = (N_NODES * 64 + 255) / 256;
  const int ginBlocks  = (TILES + 3) / 4;
  const int gBlocks    = (N_GRAPHS + 255) / 256;

  // ---- layer 1: GINConv(6 -> 64 -> 64)
  launch_zero(S.AGG6, N_NODES * 6, st);
  edge_agg6_kernel<<<(int)(((long)N_EDGES * 6 + 255) / 256), 256, 0, st>>>(x, src, dst, S.AGG6);
  gin_mlp6_kernel<<<ginBlocks, 128, 0, st>>>(x, S.AGG6, w11, b11, w12, b12, S.B0);

  // ---- graph-LN 1 + relu (also establishes per-graph node counts)
  launch_zero(S.GCNT, N_GRAPHS, st);
  launch_zero(S.GSUM, N_GRAPHS, st);
  stats1_kernel<<<nodeBlocks, 256, 0, st>>>(S.B0, batch, S.GSUM, S.GCNT, 1);
  ln_mean_kernel<<<gBlocks, 256, 0, st>>>(S.GSUM, S.GCNT, S.GMEAN);
  launch_zero(S.GVAR, N_GRAPHS, st);
  stats2_kernel<<<nodeBlocks, 256, 0, st>>>(S.B0, batch, S.GMEAN, S.GVAR);
  ln_rsig_kernel<<<gBlocks, 256, 0, st>>>(S.GVAR, S.GCNT, S.GRSIG);
  ln_apply_relu_kernel<<<ncBlocks, 256, 0, st>>>(S.B0, batch, S.GMEAN, S.GRSIG, ln1w, ln1b);

  // ---- layer 2: GINConv(64 -> 64 -> 64)
  launch_zero(S.B1, N_NODES * 64, st);
  edge_agg64_kernel<<<(int)(((long)N_EDGES * 16 + 255) / 256), 256, 0, st>>>(S.B0, src, dst, S.B1);
  gin_mlp64_kernel<<<ginBlocks, 128, 0, st>>>(S.B0, S.B1, w21, b21, w22, b22, S.B0);

  // ---- graph-LN 2 + relu
  launch_zero(S.GSUM, N_GRAPHS, st);
  stats1_kernel<<<nodeBlocks, 256, 0, st>>>(S.B0, batch, S.GSUM, S.GCNT, 0);
  ln_mean_kernel<<<gBlocks, 256, 0, st>>>(S.GSUM, S.GCNT, S.GMEAN);
  launch_zero(S.GVAR, N_GRAPHS, st);
  stats2_kernel<<<nodeBlocks, 256, 0, st>>>(S.B0, batch, S.GMEAN, S.GVAR);
  ln_rsig_kernel<<<gBlocks, 256, 0, st>>>(S.GVAR, S.GCNT, S.GRSIG);
  ln_apply_relu_kernel<<<ncBlocks, 256, 0, st>>>(S.B0, batch, S.GMEAN, S.GRSIG, ln2w, ln2b);

  // ---- global pooling: add + mean
  launch_zero(S.GADD, N_GRAPHS * 64, st);
  pool_kernel<<<ncBlocks, 256, 0, st>>>(S.B0, batch, S.GADD);
  pool_finish_kernel<<<(N_GRAPHS * 64 + 255) / 256, 256, 0, st>>>(S.GADD, S.GCNT, emb);
}
} // namespace

extern "C" void kernel_launch(void* const* d_in, const int* in_sizes, int n_in,
                              void* d_out, int out_size, void* d_ws, size_t ws_size,
                              hipStream_t stream) {
  const float*     x1  = (const float*)d_in[0];
  const long long* ei1 = (const long long*)d_in[1];
  const long long* bt1 = (const long long*)d_in[2];
  const float*     x2  = (const float*)d_in[3];
  const long long* ei2 = (const long long*)d_in[4];
  const long long* bt2 = (const long long*)d_in[5];
  const float* d1  = (const float*)d_in[6];
  const float* d2  = (const float*)d_in[7];
  const float* w11 = (const float*)d_in[8];
  const float* b11 = (const float*)d_in[9];
  const float* w12 = (const float*)d_in[10];
  const float* b12 = (const float*)d_in[11];
  const float* ln1w = (const float*)d_in[12];
  const float* ln1b = (const float*)d_in[13];
  const float* w21 = (const float*)d_in[14];
  const float* b21 = (const float*)d_in[15];
  const float* w22 = (const float*)d_in[16];
  const float* b22 = (const float*)d_in[17];
  const float* ln2w = (const float*)d_in[18];
  const float* ln2b = (const float*)d_in[19];
  const float* fc1w = (const float*)d_in[20];
  const float* fc1b = (const float*)d_in[21];
  const float* fc2w = (const float*)d_in[22];
  const float* fc2b = (const float*)d_in[23];
  const float* outw = (const float*)d_in[24];
  const float* outb = (const float*)d_in[25];
  float* out = (float*)d_out;

  // workspace carve-up (floats)
  float* W = (float*)d_ws;
  Scratch S;
  S.B0    = W;                       W += (long)N_NODES * 64;
  S.B1    = W;                       W += (long)N_NODES * 64;
  S.AGG6  = W;                       W += (long)N_NODES * 6;
  S.GCNT  = W;                       W += N_GRAPHS;
  S.GSUM  = W;                       W += N_GRAPHS;
  S.GVAR  = W;                       W += N_GRAPHS;
  S.GMEAN = W;                       W += N_GRAPHS;
  S.GRSIG = W;                       W += N_GRAPHS;
  S.GADD  = W;                       W += N_GRAPHS * 64;
  float* EMB1 = W;                   W += N_GRAPHS * 64;
  float* EMB2 = W;                   W += N_GRAPHS * 64;
  float* H1   = W;                   W += N_GRAPHS * 128;
  float* H2   = W;                   W += N_GRAPHS * 64;

  run_backbone(x1, ei1, bt1, w11, b11, w12, b12, ln1w, ln1b,
               w21, b21, w22, b22, ln2w, ln2b, EMB1, S, stream);
  run_backbone(x2, ei2, bt2, w11, b11, w12, b12, ln1w, ln1b,
               w21, b21, w22, b22, ln2w, ln2b, EMB2, S, stream);

  fc1_kernel<<<(N_GRAPHS * 128 + 255) / 256, 256, 0, stream>>>(EMB1, EMB2, d1, d2, fc1w, fc1b, H1);
  fc2_kernel<<<(N_GRAPHS * 64 + 255) / 256, 256, 0, stream>>>(H1, fc2w, fc2b, H2);
  out_kernel<<<(N_GRAPHS + 255) / 256, 256, 0, stream>>>(H2, outw, outb, out);
}